// HierarchicalHistogramLoss_37254546325525
// MI455X (gfx1250) — compile-verified
//
#include <hip/hip_runtime.h>
#include <hip/hip_bf16.h>

typedef __attribute__((ext_vector_type(2))) float v2f;
typedef __attribute__((ext_vector_type(8))) float v8f;

#define NUM_BINS 512
#define EPS 1e-8f

// ---------------------------------------------------------------------------
// Kernel A: fine soft histograms. One block (256 threads) per 32x32 patch.
// Each pixel scatters triangular-kernel weight to its two neighboring bins
// via LDS float atomics (ds_add_f32). Exactly matches
// relu(1 - |x - c_k|/bw) since only the two flanking bins can be nonzero.
// ---------------------------------------------------------------------------
__global__ void hist_fine_kernel(const float* __restrict__ gen,
                                 float* __restrict__ fine) {
    __shared__ float h[NUM_BINS];
    const int tid = threadIdx.x;
    h[tid] = 0.f;
    h[tid + 256] = 0.f;
    __syncthreads();

    const int blk   = blockIdx.x;       // b*64 + patch
    const int b     = blk >> 6;
    const int patch = blk & 63;
    const int gy    = patch >> 3, gx = patch & 7;
    const float* src = gen + (size_t)b * 256 * 256 + (size_t)(gy * 32) * 256 + gx * 32;

#pragma unroll
    for (int t = 0; t < 4; ++t) {
        const int p  = tid + t * 256;
        const int iy = p >> 5, ix = p & 31;
        const float x  = src[iy * 256 + ix];
        const float tt = (x + 1.0f) * 255.5f;     // (x - MIN)/bin_width
        const float fj = floorf(tt);
        const int   j  = (int)fj;
        const float w1 = tt - fj;                 // weight for bin j+1
        const float w0 = 1.0f - w1;               // weight for bin j
        if (j >= 0 && j < NUM_BINS && w0 > 0.f)           atomicAdd(&h[j],     w0);
        if (j + 1 >= 0 && j + 1 < NUM_BINS && w1 > 0.f)   atomicAdd(&h[j + 1], w1);
    }
    __syncthreads();
    float* dst = fine + (size_t)blk * NUM_BINS;
    dst[tid]       = h[tid];
    dst[tid + 256] = h[tid + 256];
}

// ---------------------------------------------------------------------------
// Kernel B: coarse histograms via exact f32 WMMA GEMM.
// Coarse[32pad x 512] = Agg[32 x 64] (0/1) x Fine[64 x 512], per batch.
// Rows: 0 = scale0; 1..4 = scale1 (row-major); 5..20 = scale2; >=21 pad (0).
// One wave (32 threads) per (batch, 16-bin column tile): 16 K-steps x 2 row
// tiles = 32 v_wmma_f32_16x16x4_f32 per wave. Everything exact in f32.
//
// Each lane's aggregation row is encoded ONCE as a 64-bit child-membership
// bitmask; per K-step the two needed A elements come from one 64-bit right
// shift + bit extract + int->float convert. No EXEC divergence in the loop.
// ---------------------------------------------------------------------------
__device__ __forceinline__ unsigned long long rowmask(int r) {
    // scale1 candidate (valid when 1 <= r < 5): 4x4 block of fine patches
    const int q1 = r - 1;
    const unsigned long long m1 =
        ((0x0FULL << (4 * (q1 & 1))) * 0x01010101ULL) << (32 * ((q1 >> 1) & 1));
    // scale2 candidate (valid when 5 <= r < 21): 2x2 block of fine patches
    const int q2 = r - 5;
    const unsigned long long m2 =
        ((0x03ULL << (2 * (q2 & 3))) * 0x0101ULL) << (16 * ((q2 >> 2) & 3));
    return (r == 0) ? ~0ULL : ((r < 5) ? m1 : ((r < 21) ? m2 : 0ULL));
}

__global__ void agg_wmma_kernel(const float* __restrict__ fine,
                                float* __restrict__ coarse) {
    const int lane  = threadIdx.x;          // 32 threads = 1 wave
    const int blk   = blockIdx.x;           // b*32 + ntile
    const int b     = blk >> 5;
    const int ntile = blk & 31;

    const float* F = fine + (size_t)b * 64 * NUM_BINS;
    const int col  = ntile * 16 + (lane & 15);
    const int ksub = (lane >> 4) * 2;       // lanes 16-31 hold K=2,3 of each step
    const int m    = lane & 15;             // A-matrix row within tile

    const unsigned long long mask0 = rowmask(m);        // rows 0..15
    const unsigned long long mask1 = rowmask(m + 16);   // rows 16..31

    v8f acc0 = {};  // rows 0..15
    v8f acc1 = {};  // rows 16..31

#pragma unroll
    for (int kstep = 0; kstep < 16; ++kstep) {
        const int krow = kstep * 4 + ksub;  // fine-patch index (K)
        v2f bfrag;
        bfrag[0] = F[(size_t)krow * NUM_BINS + col];
        bfrag[1] = F[(size_t)(krow + 1) * NUM_BINS + col];
        const unsigned e0 = (unsigned)(mask0 >> krow);
        const unsigned e1 = (unsigned)(mask1 >> krow);
        v2f a0, a1;
        a0[0] = (float)(e0 & 1u);
        a0[1] = (float)((e0 >> 1) & 1u);
        a1[0] = (float)(e1 & 1u);
        a1[1] = (float)((e1 >> 1) & 1u);
        acc0 = __builtin_amdgcn_wmma_f32_16x16x4_f32(false, a0, false, bfrag,
                                                     (short)0, acc0, false, false);
        acc1 = __builtin_amdgcn_wmma_f32_16x16x4_f32(false, a1, false, bfrag,
                                                     (short)0, acc1, false, false);
    }

    // D layout: VGPR v -> row v (lanes 0-15) / row v+8 (lanes 16-31).
    float* C = coarse + (size_t)b * 21 * NUM_BINS;
    const int half8 = (lane >> 4) * 8;
#pragma unroll
    for (int v = 0; v < 8; ++v) {
        const int r0 = v + half8;
        if (r0 < 21) C[(size_t)r0 * NUM_BINS + col] = acc0[v];
        const int r1 = 16 + v + half8;
        if (r1 < 21) C[(size_t)r1 * NUM_BINS + col] = acc1[v];
    }
}

// ---------------------------------------------------------------------------
// Kernel C: per-patch Wasserstein term. One wave32 per (batch, patch-any-scale).
// 16 bins/lane; wave reduction for sums, shfl_up scan for the CDF.
// Writes per-patch loss to workspace (deterministic final reduce later).
// ---------------------------------------------------------------------------
__global__ void loss_kernel(const float* __restrict__ fine,
                            const float* __restrict__ coarse,
                            const float* __restrict__ t0,
                            const float* __restrict__ t1,
                            const float* __restrict__ t2,
                            const float* __restrict__ t3,
                            float* __restrict__ losses) {
    const int lane = threadIdx.x;
    const int blk  = blockIdx.x;            // b*85 + r
    const int b    = blk / 85;
    const int r    = blk % 85;

    const float* gh;
    const float* th;
    if (r == 0)       { gh = coarse + (size_t)b * 21 * NUM_BINS;
                        th = t0 + (size_t)b * NUM_BINS; }
    else if (r < 5)   { gh = coarse + ((size_t)b * 21 + r) * NUM_BINS;
                        th = t1 + ((size_t)b * 4 + (r - 1)) * NUM_BINS; }
    else if (r < 21)  { gh = coarse + ((size_t)b * 21 + r) * NUM_BINS;
                        th = t2 + ((size_t)b * 16 + (r - 5)) * NUM_BINS; }
    else              { gh = fine + ((size_t)b * 64 + (r - 21)) * NUM_BINS;
                        th = t3 + ((size_t)b * 64 + (r - 21)) * NUM_BINS; }

    float g[16], t[16];
    float gs = 0.f, ts = 0.f;
    const int base = lane * 16;
#pragma unroll
    for (int i = 0; i < 16; ++i) {
        g[i] = gh[base + i];
        t[i] = th[base + i];
        gs += g[i]; ts += t[i];
    }
#pragma unroll
    for (int off = 16; off > 0; off >>= 1) {
        gs += __shfl_xor(gs, off, 32);
        ts += __shfl_xor(ts, off, 32);
    }
    const float gi = 1.f / (gs + EPS);
    const float ti = 1.f / (ts + EPS);

    float d[16];
    float l1 = 0.f, chunk = 0.f;
#pragma unroll
    for (int i = 0; i < 16; ++i) {
        d[i] = g[i] * gi - t[i] * ti;
        l1 += fabsf(d[i]);
        chunk += d[i];
    }
    // exclusive prefix of per-lane chunk sums across the wave
    float incl = chunk;
#pragma unroll
    for (int off = 1; off < 32; off <<= 1) {
        const float n = __shfl_up(incl, off, 32);
        if (lane >= off) incl += n;
    }
    float run = incl - chunk;   // exclusive prefix
    float cdfa = 0.f;
#pragma unroll
    for (int i = 0; i < 16; ++i) {
        run += d[i];
        cdfa += fabsf(run);
    }
#pragma unroll
    for (int off = 16; off > 0; off >>= 1) {
        l1   += __shfl_xor(l1, off, 32);
        cdfa += __shfl_xor(cdfa, off, 32);
    }
    if (lane == 0)
        losses[blk] = (l1 + cdfa) * (1.0f / (float)NUM_BINS);
}

// ---------------------------------------------------------------------------
// Kernel D: deterministic final reduction of per-patch losses.
// ---------------------------------------------------------------------------
__global__ void reduce_kernel(const float* __restrict__ losses, int n,
                              float scale, float* __restrict__ out) {
    __shared__ float s[256];
    const int tid = threadIdx.x;
    float acc = 0.f;
    for (int i = tid; i < n; i += 256) acc += losses[i];
    s[tid] = acc;
    __syncthreads();
    for (int off = 128; off > 0; off >>= 1) {
        if (tid < off) s[tid] += s[tid + off];
        __syncthreads();
    }
    if (tid == 0) out[0] = s[0] * scale;
}

extern "C" void kernel_launch(void* const* d_in, const int* in_sizes, int n_in,
                              void* d_out, int out_size, void* d_ws, size_t ws_size,
                              hipStream_t stream) {
    const float* gen = (const float*)d_in[0];
    const float* t0  = (const float*)d_in[1];
    const float* t1  = (const float*)d_in[2];
    const float* t2  = (const float*)d_in[3];
    const float* t3  = (const float*)d_in[4];
    float* out = (float*)d_out;

    const int B = in_sizes[0] / (256 * 256);

    float* fine   = (float*)d_ws;                         // B*64*512
    float* coarse = fine   + (size_t)B * 64 * NUM_BINS;   // B*21*512
    float* losses = coarse + (size_t)B * 21 * NUM_BINS;   // B*85

    hist_fine_kernel<<<B * 64, 256, 0, stream>>>(gen, fine);
    agg_wmma_kernel<<<B * 32, 32, 0, stream>>>(fine, coarse);
    loss_kernel<<<B * 85, 32, 0, stream>>>(fine, coarse, t0, t1, t2, t3, losses);
    reduce_kernel<<<1, 256, 0, stream>>>(losses, B * 85,
                                         1.0f / ((float)B * 85.0f), out);
}